// GATLayer_24043226923662
// MI455X (gfx1250) — compile-verified
//
#include <hip/hip_runtime.h>
#include <hip/hip_bf16.h>
#include <math.h>

#define IN_FEAT 128
#define OH 64   // OUT_FEAT * N_HEADS

typedef float v2f __attribute__((ext_vector_type(2)));
typedef float v8f __attribute__((ext_vector_type(8)));

// ---- order-preserving float<->uint mapping (for atomicMax on floats) ----
__device__ __forceinline__ unsigned f2ord(float f) {
  unsigned u = __float_as_uint(f);
  return (u & 0x80000000u) ? ~u : (u | 0x80000000u);
}
__device__ __forceinline__ float ord2f(unsigned u) {
  unsigned b = (u & 0x80000000u) ? (u & 0x7FFFFFFFu) : ~u;
  return __uint_as_float(b);
}

// ---- 0: reduce sum(a) (128 elems), init softmax scalars ----
__global__ void init_scalars(const float* __restrict__ a, float* sumA,
                             unsigned* maxbits, float* expsum) {
  __shared__ float red[128];
  int t = threadIdx.x;
  red[t] = a[t];
  __syncthreads();
  for (int s = 64; s > 0; s >>= 1) {
    if (t < s) red[t] += red[t + s];
    __syncthreads();
  }
  if (t == 0) {
    *sumA = red[0];
    *maxbits = f2ord(-INFINITY);
    *expsum = 0.0f;
  }
}

__global__ void zero_f32(float* __restrict__ p, int n) {
  int i = blockIdx.x * blockDim.x + threadIdx.x;
  if (i < n) p[i] = 0.0f;
}

// ---- 1: Wh = h @ W via V_WMMA_F32_16X16X4_F32 ----
// Block = 128 threads (4 waves). Block b computes rows [16b,16b+16); wave w
// computes the 16-column tile n0 = 16w. K=128 -> 32 WMMA(16x16x4) per wave.
// A 16x4 f32 layout: lanes 0-15 -> M=lane, K={0,1}; lanes 16-31 -> M=lane-16, K={2,3}.
// B 4x16 f32 layout mirrors A with N in place of M.
// C/D 16x16 f32: VGPR r -> M=r (lanes 0-15) / M=r+8 (lanes 16-31), N=lane&15.
__global__ void __launch_bounds__(128) gemm_wh_wmma(const float* __restrict__ h,
                                                    const float* __restrict__ W,
                                                    float* __restrict__ Wh,
                                                    int nNodes) {
  const int lane = threadIdx.x & 31;
  const int wave = threadIdx.x >> 5;
  const int row0 = blockIdx.x << 4;
  const int n0   = wave << 4;
  const int m    = lane & 15;
  const int koff = (lane >> 4) << 1;            // 0 for lanes 0-15, 2 for lanes 16-31

  int row = row0 + m;
  if (row >= nNodes) row = nNodes - 1;          // clamp so EXEC stays all-ones
  const float* hp = h + (size_t)row * IN_FEAT + koff;
  const float* wp = W + (size_t)koff * OH + n0 + m;

  v8f acc = {};
  for (int k = 0; k < IN_FEAT; k += 4) {
    v2f af;
    af.x = hp[k];                                // (M, k+koff)
    af.y = hp[k + 1];                            // (M, k+koff+1)
    v2f bf;
    bf.x = wp[(size_t)k * OH];                   // (k+koff,   N)
    bf.y = wp[(size_t)(k + 1) * OH];             // (k+koff+1, N)
    acc = __builtin_amdgcn_wmma_f32_16x16x4_f32(
        false, af, false, bf, (short)0, acc, false, false);
  }

  const int rbase = row0 + ((lane >> 4) << 3);
  const int col   = n0 + m;
#pragma unroll
  for (int r = 0; r < 8; ++r) {
    int rr = rbase + r;
    if (rr < nNodes) Wh[(size_t)rr * OH + col] = acc[r];
  }
}

// ---- 2: per-edge scores + global max (16 lanes per edge, float4 per lane) ----
__global__ void __launch_bounds__(1024) edge_scores(const float* __restrict__ Wh,
                                                    const long long* __restrict__ ei,
                                                    const float* __restrict__ sumA,
                                                    float* __restrict__ evals,
                                                    unsigned* __restrict__ maxbits,
                                                    int E) {
  __shared__ float smax[64];
  const int gid = (int)((blockIdx.x * (size_t)blockDim.x + threadIdx.x) >> 4);
  const int l = threadIdx.x & 15;

  float part = 0.0f;
  if (gid < E) {
    long long s = ei[gid];
    long long d = ei[(size_t)E + gid];
    float4 va = ((const float4*)(Wh + (size_t)s * OH))[l];
    float4 vb = ((const float4*)(Wh + (size_t)d * OH))[l];
    part = va.x * vb.x + va.y * vb.y + va.z * vb.z + va.w * vb.w;
  }
  part += __shfl_xor(part, 8);
  part += __shfl_xor(part, 4);
  part += __shfl_xor(part, 2);
  part += __shfl_xor(part, 1);

  float ev = -INFINITY;
  if (gid < E && l == 0) {
    float sc = part * sumA[0];
    ev = sc > 0.0f ? sc : 0.2f * sc;             // leaky_relu(0.2)
    evals[gid] = ev;
  }
  if (l == 0) smax[threadIdx.x >> 4] = ev;
  __syncthreads();
  if (threadIdx.x == 0) {
    float mx = smax[0];
    for (int i = 1; i < 64; ++i) mx = fmaxf(mx, smax[i]);
    atomicMax(maxbits, f2ord(mx));               // one contended atomic per block
  }
}

// ---- 3: sum of exp(e - max) ----
__global__ void __launch_bounds__(256) exp_sum(const float* __restrict__ evals,
                                               const unsigned* __restrict__ maxbits,
                                               float* __restrict__ expsum, int E) {
  __shared__ float red[256];
  const float gmax = ord2f(*maxbits);
  float acc = 0.0f;
  for (int i = blockIdx.x * blockDim.x + threadIdx.x; i < E;
       i += gridDim.x * blockDim.x)
    acc += expf(evals[i] - gmax);
  red[threadIdx.x] = acc;
  __syncthreads();
  for (int s = 128; s > 0; s >>= 1) {
    if (threadIdx.x < s) red[threadIdx.x] += red[threadIdx.x + s];
    __syncthreads();
  }
  if (threadIdx.x == 0) atomicAdd(expsum, red[0]);
}

__global__ void finalize_inv(const float* __restrict__ expsum,
                             float* __restrict__ invsum) {
  *invsum = 1.0f / *expsum;
}

// ---- 4: scatter h_prime[dst] += att * Wh[src] (16 lanes/edge, 4 atomics each) ----
__global__ void __launch_bounds__(1024) scatter_edges(const float* __restrict__ Wh,
                                                      const long long* __restrict__ ei,
                                                      const float* __restrict__ evals,
                                                      const unsigned* __restrict__ maxbits,
                                                      const float* __restrict__ invsum,
                                                      float* __restrict__ out, int E) {
  const int gid = (int)((blockIdx.x * (size_t)blockDim.x + threadIdx.x) >> 4);
  if (gid >= E) return;
  const int l = threadIdx.x & 15;

  const float att = expf(evals[gid] - ord2f(*maxbits)) * (*invsum);
  long long s = ei[gid];
  long long d = ei[(size_t)E + gid];
  float4 v = ((const float4*)(Wh + (size_t)s * OH))[l];
  float* op = out + (size_t)d * OH + (l << 2);
  atomicAdd(op + 0, att * v.x);
  atomicAdd(op + 1, att * v.y);
  atomicAdd(op + 2, att * v.z);
  atomicAdd(op + 3, att * v.w);
}

// ---- 5: elu in place ----
__global__ void elu_inplace(float* __restrict__ p, int n) {
  int i = blockIdx.x * blockDim.x + threadIdx.x;
  if (i < n) {
    float x = p[i];
    p[i] = x > 0.0f ? x : expm1f(x);
  }
}

extern "C" void kernel_launch(void* const* d_in, const int* in_sizes, int n_in,
                              void* d_out, int out_size, void* d_ws, size_t ws_size,
                              hipStream_t stream) {
  const float*     h  = (const float*)d_in[0];
  const long long* ei = (const long long*)d_in[1];   // int64 edge_index [2, E]
  const float*     W  = (const float*)d_in[2];
  const float*     a  = (const float*)d_in[3];

  const int nNodes = in_sizes[0] / IN_FEAT;          // 50000
  const int E      = in_sizes[1] / 2;                // 800000
  const int nOut   = nNodes * OH;                    // == out_size

  float* out = (float*)d_out;

  // workspace layout (floats): Wh | evals | sumA | maxbits | expsum | invsum
  float*    Wh      = (float*)d_ws;
  float*    evals   = Wh + (size_t)nNodes * OH;
  float*    sumA    = evals + E;
  unsigned* maxbits = (unsigned*)(sumA + 1);
  float*    expsum  = (float*)(maxbits + 1);
  float*    invsum  = expsum + 1;

  init_scalars<<<1, 128, 0, stream>>>(a, sumA, maxbits, expsum);
  zero_f32<<<(nOut + 255) / 256, 256, 0, stream>>>(out, nOut);

  gemm_wh_wmma<<<(nNodes + 15) / 16, 128, 0, stream>>>(h, W, Wh, nNodes);

  const int ethreads = E * 16;
  edge_scores<<<(ethreads + 1023) / 1024, 1024, 0, stream>>>(Wh, ei, sumA, evals,
                                                             maxbits, E);
  exp_sum<<<512, 256, 0, stream>>>(evals, maxbits, expsum, E);
  finalize_inv<<<1, 1, 0, stream>>>(expsum, invsum);
  scatter_edges<<<(ethreads + 1023) / 1024, 1024, 0, stream>>>(Wh, ei, evals,
                                                               maxbits, invsum, out, E);
  elu_inplace<<<(nOut + 255) / 256, 256, 0, stream>>>(out, nOut);
}